// MorphClassifier_27376121545074
// MI455X (gfx1250) — compile-verified
//
#include <hip/hip_runtime.h>

// WOS (weighted-offset-serial) forward, D=2, D2=4, STACK_BITS=8.
//
// Per sample: 4 lanes = [x0, x1, -x0, -x1] + bias, each serialized to 8
// offset-binary bits (MSB first). 8-step recurrence with sticky lanes and a
// thresholded weighted popcount per step. Weights/threshold are uniform, so
// the length-4 dot product + compare folds into a 16-entry LUT packed into
// one 32-bit mask M (bit c = [sum of weights over set bits of c >= thr]).
//
// Memory-bound-ish: 12 MB total traffic (fits in L2), ~0.5us HBM floor.
// Inner loop is pure bitfield VALU (v_bfe/v_bfi/v_lshl_or); WMMA is
// inapplicable (depth-8 serial recurrence, inner dot length 4 -> uniform LUT).

// Spread the 8 low bits of v to stride-4 positions: bit j -> bit 4j.
__device__ __forceinline__ unsigned spread8(unsigned v) {
    unsigned t = (v | (v << 12)) & 0x000F000Fu;   // nibbles at [3:0], [19:16]
    t = (t | (t << 6)) & 0x03030303u;             // bit-pairs at byte bases
    t = (t | (t << 3)) & 0x11111111u;             // single bits at nibble bases
    return t;
}

__device__ __forceinline__ float wos_one(float x0, float x1,
                                         float b0, float b1, float b2, float b3,
                                         unsigned M) {
    // lsb lanes = [x0+b0, x1+b1, -x0+b2, -x1+b3]; serialize: (int(v)+128)&0xFF
    // (int) truncates toward zero, matching jnp astype(int32).
    unsigned v0 = ((unsigned)((int)(x0 + b0) + 128)) & 0xFFu;
    unsigned v1 = ((unsigned)((int)(x1 + b1) + 128)) & 0xFFu;
    unsigned v2 = ((unsigned)((int)(-x0 + b2) + 128)) & 0xFFu;
    unsigned v3 = ((unsigned)((int)(-x1 + b3) + 128)) & 0xFFu;

    // B: nibble p holds the 4 lane bits at bit position p (lane j -> bit j).
    unsigned B = spread8(v0) | (spread8(v1) << 1) | (spread8(v2) << 2) | (spread8(v3) << 3);

    unsigned cur = 0u, stuck = 0u, bits = 0u;
    // Reference step k uses bit position p = 7-k (MSB first); iterate p = 7..0.
    #pragma unroll
    for (int p = 7; p >= 0; --p) {
        unsigned bk = (B >> (4 * p)) & 0xFu;            // v_bfe_u32
        cur = (stuck & cur) | (bk & ~stuck);            // v_bfi_b32: sticky lanes hold
        unsigned ob = (M >> cur) & 1u;                  // v_bfe_u32: LUT'd wsum >= thr
        unsigned obm = 0u - ob;                         // 0 or 0xFFFFFFFF
        stuck |= (bk ^ obm);                            // diverging lanes go sticky
        bits |= ob << p;                                // deserialize MSB-first
    }
    return (float)bits - 128.0f;
}

#define WOS_SPT 8  // samples per thread (amortizes LUT build; 4x b128 in, 2x b128 out)

__global__ __launch_bounds__(256) void wos_kernel(
    const float* __restrict__ x,          // [n, 2] row-major
    const float* __restrict__ biases,     // [4]
    const float* __restrict__ weights,    // [4]
    const float* __restrict__ threshold,  // [1]
    float* __restrict__ out,              // [n]
    int n) {
    const float b0 = biases[0], b1 = biases[1], b2 = biases[2], b3 = biases[3];
    const float w0 = weights[0], w1 = weights[1], w2 = weights[2], w3 = weights[3];
    const float thr = threshold[0];

    // Uniform 16-entry LUT in one 32-bit word. c is compile-time in the
    // unrolled loop, so the selects constant-fold to plain adds.
    unsigned M = 0u;
    #pragma unroll
    for (int c = 0; c < 16; ++c) {
        float s = ((c & 1) ? w0 : 0.0f) + ((c & 2) ? w1 : 0.0f) +
                  ((c & 4) ? w2 : 0.0f) + ((c & 8) ? w3 : 0.0f);
        M |= (s >= thr ? 1u : 0u) << c;
    }

    const int t  = blockIdx.x * blockDim.x + threadIdx.x;
    const int i0 = t * WOS_SPT;
    if (i0 >= n) return;

    if (i0 + WOS_SPT <= n) {
        // 8 samples = 16 input floats = 4x float4, 8 output floats = 2x float4.
        const float4* xv = (const float4*)x;
        float4 a0 = xv[4 * t + 0];
        float4 a1 = xv[4 * t + 1];
        float4 a2 = xv[4 * t + 2];
        float4 a3 = xv[4 * t + 3];
        float xs[16] = {a0.x, a0.y, a0.z, a0.w, a1.x, a1.y, a1.z, a1.w,
                        a2.x, a2.y, a2.z, a2.w, a3.x, a3.y, a3.z, a3.w};
        float os[8];
        #pragma unroll
        for (int s = 0; s < 8; ++s)
            os[s] = wos_one(xs[2 * s], xs[2 * s + 1], b0, b1, b2, b3, M);
        float4* ov = (float4*)out;
        ov[2 * t + 0] = make_float4(os[0], os[1], os[2], os[3]);
        ov[2 * t + 1] = make_float4(os[4], os[5], os[6], os[7]);
    } else {
        // Tail (n not a multiple of WOS_SPT)
        for (int i = i0; i < n; ++i)
            out[i] = wos_one(x[2 * i], x[2 * i + 1], b0, b1, b2, b3, M);
    }
}

extern "C" void kernel_launch(void* const* d_in, const int* in_sizes, int n_in,
                              void* d_out, int out_size, void* d_ws, size_t ws_size,
                              hipStream_t stream) {
    const float* x         = (const float*)d_in[0];  // [N*2]
    const float* biases    = (const float*)d_in[1];  // [4]
    const float* weights   = (const float*)d_in[2];  // [4]
    const float* threshold = (const float*)d_in[3];  // [1]
    float* out             = (float*)d_out;          // [N]

    const int n = in_sizes[0] / 2;
    const int threads = (n + WOS_SPT - 1) / WOS_SPT;
    const int block = 256;
    const int grid = (threads + block - 1) / block;
    wos_kernel<<<grid, block, 0, stream>>>(x, biases, weights, threshold, out, n);
}